// Net_32787780337936
// MI455X (gfx1250) — compile-verified
//
#include <hip/hip_runtime.h>
#include <hip/hip_bf16.h>

typedef float v2f __attribute__((ext_vector_type(2)));
typedef float v8f __attribute__((ext_vector_type(8)));

#define EPSBN 1e-5f

// ws layout (float offsets)
#define OFF_W0F   0u        // 4 ksteps * 4 ntiles * 32 lanes * 2 = 1024 floats
#define OFF_WABF  1024u     // 4 convs * (16 ksteps * 8 ntiles * 32 * 2 = 8192)
#define OFF_WLF   33792u    // 16 ksteps * 4 ntiles * 32 * 2 = 4096
#define OFF_BIAS0 37888u    // 64
#define OFF_BIASA 37952u    // 4 * 64
#define OFF_BIASL 38208u    // 64
#define OFF_BUF   38400u    // node buffers start here

__device__ __forceinline__ v8f wmma_f32(v2f a, v2f b, v8f c) {
    // D = A(16x4 f32) * B(4x16 f32) + C(16x16 f32)
    return __builtin_amdgcn_wmma_f32_16x16x4_f32(
        /*neg_a=*/false, a, /*neg_b=*/false, b,
        /*c_mod=*/(short)0, c, /*reuse_a=*/false, /*reuse_b=*/false);
}

// ---------------------------------------------------------------------------
// Fold BN into weights/bias and pack weights into WMMA B-fragment layout.
// B fragment (4x16 f32): lane l -> N = l%16, VGPR0 holds K = k0+2*(l/16),
// VGPR1 holds K = k0+2*(l/16)+1. Stored as float2 per (kstep,ntile,lane).
// ---------------------------------------------------------------------------
__global__ void prep_kernel(
    const float* __restrict__ w0, const float* __restrict__ b0,
    const float* __restrict__ g0, const float* __restrict__ be0,
    const float* __restrict__ w1, const float* __restrict__ b1,
    const float* __restrict__ g1, const float* __restrict__ be1,
    const float* __restrict__ w2, const float* __restrict__ b2,
    const float* __restrict__ g2, const float* __restrict__ be2,
    const float* __restrict__ w3, const float* __restrict__ b3,
    const float* __restrict__ g3, const float* __restrict__ be3,
    const float* __restrict__ w4, const float* __restrict__ b4,
    const float* __restrict__ g4, const float* __restrict__ be4,
    const float* __restrict__ wl, const float* __restrict__ bl,
    const float* __restrict__ gl, const float* __restrict__ bel,
    float* __restrict__ ws)
{
    const int tid = threadIdx.x;          // single block of 256
    const float inv = rsqrtf(1.0f + EPSBN);

    // fused biases: g'*b + beta
    if (tid < 64) {
        ws[OFF_BIAS0 + tid]        = g0[tid] * inv * b0[tid] + be0[tid];
        ws[OFF_BIASA + 0*64 + tid] = g1[tid] * inv * b1[tid] + be1[tid];
        ws[OFF_BIASA + 1*64 + tid] = g2[tid] * inv * b2[tid] + be2[tid];
        ws[OFF_BIASA + 2*64 + tid] = g3[tid] * inv * b3[tid] + be3[tid];
        ws[OFF_BIASA + 3*64 + tid] = g4[tid] * inv * b4[tid] + be4[tid];
        ws[OFF_BIASL + tid]        = gl[tid] * inv * bl[tid] + bel[tid];
    }

    // lin0 weights: [16,64] scaled by g0'
    for (int idx = tid; idx < 4 * 4 * 32; idx += 256) {
        int k    = idx >> 7;          // /128
        int nt   = (idx >> 5) & 3;
        int lane = idx & 31;
        int h = lane >> 4, c = lane & 15;
        int r0  = 4 * k + 2 * h;
        int col = nt * 16 + c;
        float gp = g0[col] * inv;
        ws[OFF_W0F + 2u * idx + 0] = w0[(r0 + 0) * 64 + col] * gp;
        ws[OFF_W0F + 2u * idx + 1] = w0[(r0 + 1) * 64 + col] * gp;
    }

    // conv weights: w is [128,64]; A-part = (Wt - Wb)*g', B-part = Wb*g'
    const float* wc[4] = { w1, w2, w3, w4 };
    const float* gc[4] = { g1, g2, g3, g4 };
    for (int cv = 0; cv < 4; ++cv) {
        const float* w = wc[cv];
        const float* g = gc[cv];
        for (int idx = tid; idx < 16 * 8 * 32; idx += 256) {
            int k    = idx >> 8;          // /(8*32)
            int nt   = (idx >> 5) & 7;
            int lane = idx & 31;
            int h = lane >> 4, c = lane & 15;
            int r0 = 4 * k + 2 * h;
            int jj = nt * 16 + c;         // 0..127 (A cols 0..63, B cols 64..127)
            for (int t = 0; t < 2; ++t) {
                int r = r0 + t;
                float val;
                if (jj < 64) {
                    float gp = g[jj] * inv;
                    val = (w[r * 64 + jj] - w[(64 + r) * 64 + jj]) * gp;
                } else {
                    int j = jj - 64;
                    float gp = g[j] * inv;
                    val = w[(64 + r) * 64 + j] * gp;
                }
                ws[OFF_WABF + (size_t)cv * 8192u + 2u * idx + t] = val;
            }
        }
    }

    // lin1 weights: [64,64] scaled by gl'
    for (int idx = tid; idx < 16 * 4 * 32; idx += 256) {
        int k    = idx >> 7;
        int nt   = (idx >> 5) & 3;
        int lane = idx & 31;
        int h = lane >> 4, c = lane & 15;
        int r0  = 4 * k + 2 * h;
        int col = nt * 16 + c;
        float gp = gl[col] * inv;
        ws[OFF_WLF + 2u * idx + 0] = wl[(r0 + 0) * 64 + col] * gp;
        ws[OFF_WLF + 2u * idx + 1] = wl[(r0 + 1) * 64 + col] * gp;
    }
}

// ---------------------------------------------------------------------------
// lin0: out[N,64] = ReLU(x[N,16] @ W0' + bias0)  (WMMA f32, K=16)
// ---------------------------------------------------------------------------
__global__ __launch_bounds__(256) void lin0_kernel(
    const float* __restrict__ x, const float* __restrict__ wf,
    const float* __restrict__ bias, float* __restrict__ out, int nTiles)
{
    int wv = threadIdx.x >> 5, lane = threadIdx.x & 31;
    int tile = blockIdx.x * 8 + wv;
    if (tile >= nTiles) return;                 // wave-uniform guard
    int h = lane >> 4, c = lane & 15;
    size_t base = (size_t)tile * 16;

    v8f acc[4] = {};
    #pragma unroll
    for (int k = 0; k < 4; ++k) {
        v2f a = *(const v2f*)(x + (base + c) * 16 + 4 * k + 2 * h);
        #pragma unroll
        for (int nt = 0; nt < 4; ++nt) {
            v2f b = *(const v2f*)(wf + ((k * 4 + nt) * 32 + lane) * 2);
            acc[nt] = wmma_f32(a, b, acc[nt]);
        }
    }
    #pragma unroll
    for (int nt = 0; nt < 4; ++nt) {
        float bv = bias[nt * 16 + c];
        #pragma unroll
        for (int v = 0; v < 8; ++v) {
            out[(base + v + 8 * h) * 64 + nt * 16 + c] = fmaxf(acc[nt][v] + bv, 0.0f);
        }
    }
}

// ---------------------------------------------------------------------------
// nodeAB: AB[N,128] = [ X@Wa + biasA | X@Wb ]   (fused 64x128 WMMA GEMM)
// ---------------------------------------------------------------------------
__global__ __launch_bounds__(256) void nodeab_kernel(
    const float* __restrict__ X, const float* __restrict__ wf,
    const float* __restrict__ biasA, float* __restrict__ AB, int nTiles)
{
    int wv = threadIdx.x >> 5, lane = threadIdx.x & 31;
    int tile = blockIdx.x * 8 + wv;
    if (tile >= nTiles) return;
    int h = lane >> 4, c = lane & 15;
    size_t base = (size_t)tile * 16;

    v8f acc[8] = {};
    #pragma unroll
    for (int k = 0; k < 16; ++k) {
        v2f a = *(const v2f*)(X + (base + c) * 64 + 4 * k + 2 * h);
        #pragma unroll
        for (int nt = 0; nt < 8; ++nt) {
            v2f b = *(const v2f*)(wf + ((k * 8 + nt) * 32 + lane) * 2);
            acc[nt] = wmma_f32(a, b, acc[nt]);
        }
    }
    #pragma unroll
    for (int nt = 0; nt < 8; ++nt) {
        float bv = (nt < 4) ? biasA[nt * 16 + c] : 0.0f;
        #pragma unroll
        for (int v = 0; v < 8; ++v) {
            AB[(base + v + 8 * h) * 128 + nt * 16 + c] = acc[nt][v] + bv;
        }
    }
}

// ---------------------------------------------------------------------------
// zero / edge / combine
// ---------------------------------------------------------------------------
__global__ void zero_kernel(float4* __restrict__ p, int n4)
{
    int i = blockIdx.x * blockDim.x + threadIdx.x;
    if (i < n4) p[i] = make_float4(0.f, 0.f, 0.f, 0.f);
}

// per edge: m = max(0, A[dst] + B[src]); agg[dst] = max(agg[dst], m)
// messages are >= 0 so int atomicMax on float bits is exact; agg pre-zeroed.
__global__ __launch_bounds__(256) void edge_kernel(
    const float* __restrict__ AB, const int* __restrict__ src,
    const int* __restrict__ dst, int* __restrict__ agg, int nE)
{
    int tid = blockIdx.x * blockDim.x + threadIdx.x;
    int e = tid >> 4;
    if (e >= nE) return;
    int q = tid & 15;
    int d = dst[e];
    int s = src[e];
    float4 a = *(const float4*)(AB + (size_t)d * 128 + q * 4);
    float4 b = *(const float4*)(AB + (size_t)s * 128 + 64 + q * 4);
    float4 m;
    m.x = fmaxf(a.x + b.x, 0.0f);
    m.y = fmaxf(a.y + b.y, 0.0f);
    m.z = fmaxf(a.z + b.z, 0.0f);
    m.w = fmaxf(a.w + b.w, 0.0f);
    int* o = agg + (size_t)d * 64 + q * 4;
    atomicMax(o + 0, __float_as_int(m.x));
    atomicMax(o + 1, __float_as_int(m.y));
    atomicMax(o + 2, __float_as_int(m.z));
    atomicMax(o + 3, __float_as_int(m.w));
}

// residual=1: xn = xp - agg (h=1.0);  residual=0: xn = agg
__global__ void combine_kernel(
    const float4* __restrict__ xp, const float4* __restrict__ agg,
    float4* __restrict__ xn, int n4, int residual)
{
    int i = blockIdx.x * blockDim.x + threadIdx.x;
    if (i >= n4) return;
    float4 g = agg[i];
    if (residual) {
        float4 xv = xp[i];
        xn[i] = make_float4(xv.x - g.x, xv.y - g.y, xv.z - g.z, xv.w - g.w);
    } else {
        xn[i] = g;
    }
}

// ---------------------------------------------------------------------------
// lin1 + log_softmax: out = logsoftmax(ReLU(X @ Wl' + biasl), axis=1)
// ---------------------------------------------------------------------------
__global__ __launch_bounds__(256) void lin1_kernel(
    const float* __restrict__ X, const float* __restrict__ wf,
    const float* __restrict__ bias, float* __restrict__ out, int nTiles)
{
    __shared__ float sm[8][16 * 66];   // per-wave staging, stride 66 -> no bank conflicts
    int wv = threadIdx.x >> 5, lane = threadIdx.x & 31;
    int tile = blockIdx.x * 8 + wv;
    if (tile >= nTiles) return;
    int h = lane >> 4, c = lane & 15;
    size_t base = (size_t)tile * 16;

    v8f acc[4] = {};
    #pragma unroll
    for (int k = 0; k < 16; ++k) {
        v2f a = *(const v2f*)(X + (base + c) * 64 + 4 * k + 2 * h);
        #pragma unroll
        for (int nt = 0; nt < 4; ++nt) {
            v2f b = *(const v2f*)(wf + ((k * 4 + nt) * 32 + lane) * 2);
            acc[nt] = wmma_f32(a, b, acc[nt]);
        }
    }
    float* ls = &sm[wv][0];
    #pragma unroll
    for (int nt = 0; nt < 4; ++nt) {
        float bv = bias[nt * 16 + c];
        #pragma unroll
        for (int v = 0; v < 8; ++v) {
            ls[(v + 8 * h) * 66 + nt * 16 + c] = fmaxf(acc[nt][v] + bv, 0.0f);
        }
    }
    // LDS ops within a wave are processed in order -> wave-local RAW is safe.
    int m   = lane >> 1;                 // node within tile
    int off = (lane & 1) * 32;           // half of the 64 channels
    float vals[32];
    float mx = 0.0f;                     // ReLU output >= 0
    #pragma unroll
    for (int j = 0; j < 32; ++j) {
        vals[j] = ls[m * 66 + off + j];
        mx = fmaxf(mx, vals[j]);
    }
    mx = fmaxf(mx, __shfl_xor(mx, 1, 32));
    float s = 0.0f;
    #pragma unroll
    for (int j = 0; j < 32; ++j) s += __expf(vals[j] - mx);
    s += __shfl_xor(s, 1, 32);
    float lz = mx + __logf(s);
    #pragma unroll
    for (int j = 0; j < 32; ++j) out[(base + m) * 64 + off + j] = vals[j] - lz;
}

// ---------------------------------------------------------------------------
extern "C" void kernel_launch(void* const* d_in, const int* in_sizes, int n_in,
                              void* d_out, int out_size, void* d_ws, size_t ws_size,
                              hipStream_t stream) {
    const float* x  = (const float*)d_in[0];
    const int*   ei = (const int*)d_in[1];
    const int N = in_sizes[0] / 16;
    const int E = in_sizes[1] / 2;
    const int* srcp = ei;
    const int* dstp = ei + E;

    const float* W[6];  const float* B[6];  const float* G[6];  const float* BE[6];
    for (int i = 0; i < 6; ++i) {
        W[i]  = (const float*)d_in[2 + 4 * i + 0];
        B[i]  = (const float*)d_in[2 + 4 * i + 1];
        G[i]  = (const float*)d_in[2 + 4 * i + 2];
        BE[i] = (const float*)d_in[2 + 4 * i + 3];
    }

    float* ws = (float*)d_ws;
    const size_t NB = (size_t)N * 64;
    float* P0  = ws + OFF_BUF;
    float* P1  = P0 + NB;
    float* AGG = P1 + NB;
    float* ABb = AGG + NB;          // [N,128]

    const int nTiles = (N + 15) / 16;
    const int gTiles = (nTiles + 7) / 8;
    const int n4     = (int)(NB / 4);
    const int gElem  = (n4 + 255) / 256;
    const int gEdge  = (E * 16 + 255) / 256;

    prep_kernel<<<1, 256, 0, stream>>>(
        W[0], B[0], G[0], BE[0], W[1], B[1], G[1], BE[1],
        W[2], B[2], G[2], BE[2], W[3], B[3], G[3], BE[3],
        W[4], B[4], G[4], BE[4], W[5], B[5], G[5], BE[5], ws);

    lin0_kernel<<<gTiles, 256, 0, stream>>>(x, ws + OFF_W0F, ws + OFF_BIAS0, P0, nTiles);

    float* Xin = P0;
    float* Xout = P1;
    const int residual[4] = { 1, 0, 1, 1 };
    for (int cv = 0; cv < 4; ++cv) {
        nodeab_kernel<<<gTiles, 256, 0, stream>>>(
            Xin, ws + OFF_WABF + (size_t)cv * 8192u, ws + OFF_BIASA + cv * 64, ABb, nTiles);
        zero_kernel<<<gElem, 256, 0, stream>>>((float4*)AGG, n4);
        edge_kernel<<<gEdge, 256, 0, stream>>>(ABb, srcp, dstp, (int*)AGG, E);
        combine_kernel<<<gElem, 256, 0, stream>>>(
            (const float4*)Xin, (const float4*)AGG, (float4*)Xout, n4, residual[cv]);
        float* t = Xin; Xin = Xout; Xout = t;
    }

    lin1_kernel<<<gTiles, 256, 0, stream>>>(Xin, ws + OFF_WLF, ws + OFF_BIASL,
                                            (float*)d_out, nTiles);
}